// EXPSAGEConv_74277164417729
// MI455X (gfx1250) — compile-verified
//
#include <hip/hip_runtime.h>

typedef __attribute__((ext_vector_type(2))) float v2f;
typedef __attribute__((ext_vector_type(8))) float v8f;

#define D        128
#define NUM_DST  25000
#define NUM_VIR  250000
#define NUM_SRC  500000
#define E0_N     1000000
#define E1_N     500000

// -------------------------------------------------------------------------
// Y[r][n] = sum_k X[r][k] * W[n][k]  (+ bias[n])   (i.e. Y = X @ W^T)
// One block = 128 threads = 4 waves; block handles a 16-row slab, each wave
// computes two 16x16 output tiles (32 output columns) via V_WMMA_F32_16X16X4.
// -------------------------------------------------------------------------
__global__ __launch_bounds__(128)
void gemm_xwt_wmma(const float* __restrict__ X, const float* __restrict__ W,
                   const float* __restrict__ bias, float* __restrict__ Y,
                   int rows)
{
    __shared__ float As[16 * 132];   // 16 x 128 A tile, padded stride 132
    const int tid      = threadIdx.x;
    const int row_base = blockIdx.x * 16;

    // Cooperative load of A tile (clamped rows for the tail slab).
    for (int i = tid; i < 16 * 32; i += 128) {
        int r  = i >> 5;
        int c4 = (i & 31) << 2;
        int gr = row_base + r;
        if (gr >= rows) gr = rows - 1;
        const float4 v = *(const float4*)(X + (size_t)gr * D + c4);
        *(float4*)(&As[r * 132 + c4]) = v;
    }
    __syncthreads();

    const int wave = tid >> 5;
    const int lane = tid & 31;
    const int m    = lane & 15;
    const int ks   = (lane >> 4) << 1;    // lanes 0-15: K={0,1}; 16-31: K={2,3}
    const int n0   = wave * 32;

    const float* arow  = &As[m * 132 + ks];
    const float* brow0 = W + (size_t)(n0 + m) * D + ks;        // B[k][n] = W[n][k]
    const float* brow1 = W + (size_t)(n0 + 16 + m) * D + ks;

    v8f c0 = {};
    v8f c1 = {};
#pragma unroll
    for (int k0 = 0; k0 < D; k0 += 4) {
        v2f a  = *(const v2f*)(arow + k0);    // ds_load_b64
        v2f b0 = *(const v2f*)(brow0 + k0);   // global_load_b64 (L2-resident W)
        v2f b1 = *(const v2f*)(brow1 + k0);
        c0 = __builtin_amdgcn_wmma_f32_16x16x4_f32(false, a, false, b0,
                                                   (short)0, c0, false, false);
        c1 = __builtin_amdgcn_wmma_f32_16x16x4_f32(false, a, false, b1,
                                                   (short)0, c1, false, false);
    }

    // C/D layout: VGPR j -> M = j (lanes 0-15) or 8+j (lanes 16-31); N = lane&15.
    const int ncol = lane & 15;
    const int moff = (lane >> 4) << 3;
    const float bb0 = bias ? bias[n0 + ncol]      : 0.0f;
    const float bb1 = bias ? bias[n0 + 16 + ncol] : 0.0f;
#pragma unroll
    for (int j = 0; j < 8; ++j) {
        int gr = row_base + moff + j;
        if (gr < rows) {
            Y[(size_t)gr * D + n0 + ncol]      = c0[j] + bb0;
            Y[(size_t)gr * D + n0 + 16 + ncol] = c1[j] + bb1;
        }
    }
}

// -------------------------------------------------------------------------
// Edge-parallel scatter-add: 32 threads per edge, float4 chunk per thread.
// lane 0 of each edge also bumps the (float) in-degree counter.
// -------------------------------------------------------------------------
__global__ __launch_bounds__(256)
void scatter_add_k(const float* __restrict__ src_feat,
                   const int* __restrict__ esrc,
                   const int* __restrict__ edst,
                   float* __restrict__ accum,
                   float* __restrict__ cnt,
                   int nedges)
{
    const long long gid = (long long)blockIdx.x * blockDim.x + threadIdx.x;
    const int e = (int)(gid >> 5);
    if (e >= nedges) return;
    const int part = (int)(gid & 31);
    const int s = esrc[e];
    const int d = edst[e];
    const float4 v = *(const float4*)(src_feat + (size_t)s * D + part * 4);
    float* o = accum + (size_t)d * D + part * 4;
    atomicAdd(o + 0, v.x);
    atomicAdd(o + 1, v.y);
    atomicAdd(o + 2, v.z);
    atomicAdd(o + 3, v.w);
    if (part == 0) atomicAdd(&cnt[d], 1.0f);
}

// accum[i] /= max(cnt[row],1)  (in place) -> turns sums into means
__global__ __launch_bounds__(256)
void seg_norm_k(float* __restrict__ accum, const float* __restrict__ cnt, size_t n)
{
    size_t i = (size_t)blockIdx.x * blockDim.x + threadIdx.x;
    if (i >= n) return;
    const float c = cnt[i >> 7];
    accum[i] *= __frcp_rn(fmaxf(c, 1.0f));
}

// out[i] += accum[i] / max(cnt[row],1)
__global__ __launch_bounds__(256)
void finalize_k(float* __restrict__ out, const float* __restrict__ accum,
                const float* __restrict__ cnt, size_t n)
{
    size_t i = (size_t)blockIdx.x * blockDim.x + threadIdx.x;
    if (i >= n) return;
    const float c = cnt[i >> 7];
    out[i] += accum[i] * __frcp_rn(fmaxf(c, 1.0f));
}

__global__ __launch_bounds__(256)
void zero_f32_k(float* __restrict__ p, size_t n)
{
    size_t i = (size_t)blockIdx.x * blockDim.x + threadIdx.x;
    if (i < n) p[i] = 0.0f;
}

// -------------------------------------------------------------------------
extern "C" void kernel_launch(void* const* d_in, const int* in_sizes, int n_in,
                              void* d_out, int out_size, void* d_ws, size_t ws_size,
                              hipStream_t stream)
{
    const float* feat    = (const float*)d_in[0];
    const int*   e0_src  = (const int*)d_in[1];
    const int*   e0_dst  = (const int*)d_in[2];
    const int*   e1_src  = (const int*)d_in[3];
    const int*   e1_dst  = (const int*)d_in[4];
    // d_in[5] = num_dst scalar (fixed module constant NUM_DST, compiled in)
    const float* W_neigh = (const float*)d_in[6];
    const float* W_self  = (const float*)d_in[7];
    const float* b_self  = (const float*)d_in[8];
    float* out = (float*)d_out;

    // Workspace layout (floats)
    float* ws    = (float*)d_ws;
    float* h_src = ws;                              // NUM_SRC * D   (256 MB)
    float* acc0  = h_src + (size_t)NUM_SRC * D;     // NUM_VIR * D   (128 MB)
    float* cnt0  = acc0  + (size_t)NUM_VIR * D;     // NUM_VIR
    float* acc1  = cnt0  + (size_t)NUM_VIR;         // NUM_DST * D   (12.8 MB)
    float* cnt1  = acc1  + (size_t)NUM_DST * D;     // NUM_DST

    // 1) zero accumulators + degree counters (contiguous from acc0)
    const size_t nz = (size_t)NUM_VIR * D + NUM_VIR + (size_t)NUM_DST * D + NUM_DST;
    zero_f32_k<<<(unsigned)((nz + 255) / 256), 256, 0, stream>>>(acc0, nz);

    // 2) h_dst = feat[:25k] @ W_self^T + b  -> d_out
    gemm_xwt_wmma<<<(NUM_DST + 15) / 16, 128, 0, stream>>>(
        feat, W_self, b_self, out, NUM_DST);

    // 3) h_src = feat[25k:] @ W_neigh^T -> workspace
    gemm_xwt_wmma<<<(NUM_SRC + 15) / 16, 128, 0, stream>>>(
        feat + (size_t)NUM_DST * D, W_neigh, nullptr, h_src, NUM_SRC);

    // 4) level-0 scatter-sum + degree
    scatter_add_k<<<(unsigned)(((long long)E0_N * 32 + 255) / 256), 256, 0, stream>>>(
        h_src, e0_src, e0_dst, acc0, cnt0, E0_N);

    // 5) sums -> means (h_vir, in place)
    seg_norm_k<<<(unsigned)(((size_t)NUM_VIR * D + 255) / 256), 256, 0, stream>>>(
        acc0, cnt0, (size_t)NUM_VIR * D);

    // 6) level-1 scatter-sum + degree
    scatter_add_k<<<(unsigned)(((long long)E1_N * 32 + 255) / 256), 256, 0, stream>>>(
        acc0, e1_src, e1_dst, acc1, cnt1, E1_N);

    // 7) out += mean(acc1)
    finalize_k<<<(unsigned)(((size_t)NUM_DST * D + 255) / 256), 256, 0, stream>>>(
        out, acc1, cnt1, (size_t)NUM_DST * D);
}